// Transformer_Attention_53635551592812
// MI455X (gfx1250) — compile-verified
//
#include <hip/hip_runtime.h>

// Transformer attention with KV cache on gfx1250 (MI455X), fp32 end-to-end.
// Bandwidth-bound (~135 MB of KV cache traffic); all GEMMs use
// V_WMMA_F32_16X16X4_F32 (wave32, 16x16 C tile per wave).

typedef __attribute__((ext_vector_type(2))) float v2f;
typedef __attribute__((ext_vector_type(8))) float v8f;

#define DIMX   512
#define SEQ    16
#define NB     8
#define KVLEN  4112
#define KTILES 257          // KVLEN / 16 ; tile 256 == freshly projected K/V
#define MAXSEQ 8192
#define QKSCALE 0.0625f     // (512/2)^-0.5

__device__ __forceinline__ v8f wmma4(v2f a, v2f b, v8f c) {
  // (neg_a, A, neg_b, B, c_mod, C, reuse_a, reuse_b)
  return __builtin_amdgcn_wmma_f32_16x16x4_f32(false, a, false, b,
                                               (short)0, c, false, false);
}

// ---------------------------------------------------------------------------
// 1) q/k/v projection: D(128,512) = X(128,512) @ W(512,512)^T for 3 weights.
//    768 tiles (mat,Mt,Nt); 4 waves / block; B[k][n] = W[n][k] -> float2 loads.
__global__ void __launch_bounds__(128)
qkv_kernel(const float* __restrict__ x,
           const float* __restrict__ Wq, const float* __restrict__ Wk,
           const float* __restrict__ Wv,
           float* __restrict__ q, float* __restrict__ kn, float* __restrict__ vn) {
  const int t    = blockIdx.x * 4 + (threadIdx.x >> 5);   // 0..767
  const int lane = threadIdx.x & 31;
  const int half = lane >> 4, lm = lane & 15;
  const int mat  = t >> 8;          // 0:q 1:k 2:v
  const int rem  = t & 255;
  const int Mt   = rem >> 5, Nt = rem & 31;

  const float* W = (mat == 0) ? Wq : (mat == 1) ? Wk : Wv;
  float*       D = (mat == 0) ? q  : (mat == 1) ? kn : vn;

  const float* arow = x + (Mt * 16 + lm) * DIMX + 2 * half;
  const float* wrow = W + (Nt * 16 + lm) * DIMX + 2 * half;

  v8f c = {};
#pragma unroll 4
  for (int k = 0; k < DIMX; k += 4) {
    v2f a = *(const v2f*)(arow + k);
    v2f b = *(const v2f*)(wrow + k);
    c = wmma4(a, b, c);
  }
  float* drow = D + Nt * 16 + lm;
#pragma unroll
  for (int i = 0; i < 8; ++i)
    drow[(Mt * 16 + i + 8 * half) * DIMX] = c[i];
}

// ---------------------------------------------------------------------------
// 2) dist = q @ k_full^T * scale  -> weight region of d_out.
//    2056 tiles = 8 batches x 257 k-tiles; tile 256 uses fresh K.
__global__ void __launch_bounds__(128)
dist_kernel(const float* __restrict__ q,
            const float* __restrict__ cache_k,
            const float* __restrict__ kn,
            float* __restrict__ wout) {
  const int t    = blockIdx.x * 4 + (threadIdx.x >> 5);   // 0..2055 exact
  const int lane = threadIdx.x & 31;
  const int half = lane >> 4, lm = lane & 15;
  const int b  = t / KTILES;
  const int kt = t - b * KTILES;

  const float* arow = q + (size_t)(b * SEQ + lm) * DIMX + 2 * half;
  const float* key  = (kt < 256)
      ? cache_k + (size_t)(b * MAXSEQ + kt * 16) * DIMX
      : kn      + (size_t)b * SEQ * DIMX;
  const float* brow = key + (size_t)lm * DIMX + 2 * half;

  v8f c = {};
#pragma unroll 4
  for (int k = 0; k < DIMX; k += 4) {
    v2f a  = *(const v2f*)(arow + k);
    v2f bb = *(const v2f*)(brow + k);
    c = wmma4(a, bb, c);
  }
  float* drow = wout + (size_t)b * SEQ * KVLEN + kt * 16 + lm;
#pragma unroll
  for (int i = 0; i < 8; ++i)
    drow[(size_t)(i + 8 * half) * KVLEN] = c[i] * QKSCALE;
}

// ---------------------------------------------------------------------------
// 3) row softmax over KVLEN with mask; probs -> ws. One block per (b,m) row.
__global__ void __launch_bounds__(256)
softmax_kernel(const float* __restrict__ wout,
               const float* __restrict__ mask,
               float* __restrict__ probs) {
  const int row = blockIdx.x;           // 0..127
  const int m   = row & 15;
  const float* drow = wout  + (size_t)row * KVLEN;
  const float* mrow = mask  + (size_t)m   * KVLEN;
  float*       prow = probs + (size_t)row * KVLEN;

  __shared__ float red[256];
  const int tid = threadIdx.x;

  float lmax = -3.402823466e38f;
  for (int j = tid; j < KVLEN; j += 256)
    lmax = fmaxf(lmax, drow[j] + mrow[j]);
  red[tid] = lmax; __syncthreads();
  for (int s = 128; s > 0; s >>= 1) {
    if (tid < s) red[tid] = fmaxf(red[tid], red[tid + s]);
    __syncthreads();
  }
  const float rmax = red[0]; __syncthreads();

  float lsum = 0.f;
  for (int j = tid; j < KVLEN; j += 256)
    lsum += __expf(drow[j] + mrow[j] - rmax);
  red[tid] = lsum; __syncthreads();
  for (int s = 128; s > 0; s >>= 1) {
    if (tid < s) red[tid] += red[tid + s];
    __syncthreads();
  }
  const float inv = 1.f / red[0];

  for (int j = tid; j < KVLEN; j += 256)
    prow[j] = __expf(drow[j] + mrow[j] - rmax) * inv;
}

// ---------------------------------------------------------------------------
// 4) atten = probs(16,KVLEN) @ v_full(KVLEN,512). Block = (b, Nt), 8 waves
//    split the 257 k-tiles (2048 waves streaming cache_v for max MLP on the
//    strided B loads); partials combined through LDS.
__global__ void __launch_bounds__(256)
atten_kernel(const float* __restrict__ probs,
             const float* __restrict__ cache_v,
             const float* __restrict__ vn,
             float* __restrict__ atten) {
  const int blk  = blockIdx.x;          // 0..255
  const int b    = blk >> 5;
  const int Nt   = blk & 31;
  const int wv   = threadIdx.x >> 5;    // 0..7
  const int lane = threadIdx.x & 31;
  const int half = lane >> 4, lm = lane & 15;

  const float* prow = probs + (size_t)(b * SEQ + lm) * KVLEN + 2 * half;

  v8f c = {};
  for (int kt = wv; kt < KTILES; kt += 8) {
    const float* vt = (kt < 256)
        ? cache_v + (size_t)(b * MAXSEQ + kt * 16) * DIMX
        : vn      + (size_t)b * SEQ * DIMX;
    const float* vcol = vt + Nt * 16 + lm;
    const int kb = kt * 16;
#pragma unroll
    for (int s = 0; s < 4; ++s) {
      v2f a = *(const v2f*)(prow + kb + 4 * s);
      v2f bb;
      bb.x = vcol[(size_t)(4 * s + 2 * half)     * DIMX];
      bb.y = vcol[(size_t)(4 * s + 2 * half + 1) * DIMX];
      c = wmma4(a, bb, c);
    }
  }

  __shared__ float red[8 * 256];
#pragma unroll
  for (int i = 0; i < 8; ++i) red[wv * 256 + i * 32 + lane] = c[i];
  __syncthreads();
  if (wv == 0) {
    #pragma unroll
    for (int i = 0; i < 8; ++i) {
      float s = 0.f;
      #pragma unroll
      for (int w = 0; w < 8; ++w) s += red[w * 256 + i * 32 + lane];
      atten[(size_t)(b * SEQ + i + 8 * half) * DIMX + Nt * 16 + lm] = s;
    }
  }
}

// ---------------------------------------------------------------------------
// 5) out = atten(128,512) @ Wo(512,512)^T  ('bqv,ev->bqe')
__global__ void __launch_bounds__(128)
outproj_kernel(const float* __restrict__ atten,
               const float* __restrict__ Wo,
               float* __restrict__ out) {
  const int t    = blockIdx.x * 4 + (threadIdx.x >> 5);   // 0..255
  const int lane = threadIdx.x & 31;
  const int half = lane >> 4, lm = lane & 15;
  const int Mt   = t >> 5, Nt = t & 31;

  const float* arow = atten + (Mt * 16 + lm) * DIMX + 2 * half;
  const float* wrow = Wo    + (Nt * 16 + lm) * DIMX + 2 * half;

  v8f c = {};
#pragma unroll 4
  for (int k = 0; k < DIMX; k += 4) {
    v2f a = *(const v2f*)(arow + k);
    v2f b = *(const v2f*)(wrow + k);
    c = wmma4(a, b, c);
  }
#pragma unroll
  for (int i = 0; i < 8; ++i)
    out[(size_t)(Mt * 16 + i + 8 * half) * DIMX + Nt * 16 + lm] = c[i];
}

// ---------------------------------------------------------------------------
extern "C" void kernel_launch(void* const* d_in, const int* in_sizes, int n_in,
                              void* d_out, int out_size, void* d_ws, size_t ws_size,
                              hipStream_t stream) {
  (void)in_sizes; (void)n_in; (void)out_size; (void)ws_size;
  // setup_inputs order: x, start_pos, mask, Wq, Wk, Wv, Wo, cache_k, cache_v
  const float* x       = (const float*)d_in[0];
  const float* mask    = (const float*)d_in[2];
  const float* Wq      = (const float*)d_in[3];
  const float* Wk      = (const float*)d_in[4];
  const float* Wv      = (const float*)d_in[5];
  const float* Wo      = (const float*)d_in[6];
  const float* cache_k = (const float*)d_in[7];
  const float* cache_v = (const float*)d_in[8];

  float* out  = (float*)d_out;                 // (8,16,512)
  float* wout = out + NB * SEQ * DIMX;         // weight (8,16,4112)

  float* ws    = (float*)d_ws;
  float* q     = ws;                                    // 65536
  float* kn    = q  + NB * SEQ * DIMX;                  // 65536
  float* vn    = kn + NB * SEQ * DIMX;                  // 65536
  float* probs = vn + NB * SEQ * DIMX;                  // 526336
  float* atten = probs + (size_t)NB * SEQ * KVLEN;      // 65536

  qkv_kernel    <<<192, 128, 0, stream>>>(x, Wq, Wk, Wv, q, kn, vn);
  dist_kernel   <<<514, 128, 0, stream>>>(q, cache_k, kn, wout);
  softmax_kernel<<<128, 256, 0, stream>>>(wout, mask, probs);
  atten_kernel  <<<256, 256, 0, stream>>>(probs, cache_v, vn, atten);
  outproj_kernel<<< 64, 128, 0, stream>>>(atten, Wo, out);
}